// ContrastiveLoss_58299886076546
// MI455X (gfx1250) — compile-verified
//
#include <hip/hip_runtime.h>

// ---------------------------------------------------------------------------
// SimCLR NT-Xent loss, fused for MI455X (gfx1250, wave32, WMMA).
//   loss = (1/2B) * sum_i [ logsumexp_{j!=i}(sim[i,j]/tau) - sim[i,(i+B)%2B]/tau ]
// Pipeline:
//   1) nrm_kernel:    z -> zn (L2-normalized rows) in workspace (16 MB, L2-resident)
//   2) simclr_kernel: per 16-row strip, async-DMA the strip into LDS
//                     (GLOBAL_LOAD_ASYNC_TO_LDS_B128), stream 16x16 sim tiles via
//                     V_WMMA_F32_16X16X4_F32 (2 column blocks / wave pass, 4
//                     independent accumulator chains) and fold into an online
//                     logsumexp — the 64 MB sim matrix never hits memory
//   3) reduce_kernel: sum 4096 row losses -> d_out[0]
// ---------------------------------------------------------------------------

#define BATCH   2048
#define NROW    4096          // 2*B
#define DIM     1024
#define TAU     0.5f
#define EPS     1e-8f
#define NBLK    256           // NROW / 16 column blocks
#define NEG_BIG (-1.0e30f)

typedef __attribute__((ext_vector_type(2))) float v2f;
typedef __attribute__((ext_vector_type(8))) float v8f;

// ---------------------------------------------------------------- normalize
__global__ __launch_bounds__(256) void nrm_kernel(const float* __restrict__ z_orig,
                                                  const float* __restrict__ z_aug,
                                                  float* __restrict__ zn) {
  __shared__ float red[8];
  const int row = blockIdx.x;
  const float* src = (row < BATCH) ? (z_aug + (size_t)row * DIM)
                                   : (z_orig + (size_t)(row - BATCH) * DIM);
  const int tid = threadIdx.x;
  float4 v = ((const float4*)src)[tid];            // 256 threads * 4 = 1024
  float ss = v.x * v.x + v.y * v.y + v.z * v.z + v.w * v.w;
  #pragma unroll
  for (int off = 16; off >= 1; off >>= 1) ss += __shfl_xor(ss, off, 32);
  if ((tid & 31) == 0) red[tid >> 5] = ss;
  __syncthreads();
  if (tid == 0) {
    float t = 0.f;
    #pragma unroll
    for (int w = 0; w < 8; ++w) t += red[w];
    red[0] = 1.0f / fmaxf(sqrtf(t), EPS);          // zn = z / max(||z||, eps)
  }
  __syncthreads();
  const float s = red[0];
  float4 o; o.x = v.x * s; o.y = v.y * s; o.z = v.z * s; o.w = v.w * s;
  ((float4*)(zn + (size_t)row * DIM))[tid] = o;
}

// ------------------------------------------------------- online-LSE update
__device__ __forceinline__ void tile_update(int cb, int rb, int pcb, float invTau,
                                            const v8f& accA, const v8f& accB,
                                            int l16, int half, float* posLds,
                                            float m[8], float l[8]) {
  #pragma unroll
  for (int v = 0; v < 8; ++v) {
    const int rloc = v + 8 * half;                 // C layout: M = v + 8*(lane>=16)
    float s = (accA[v] + accB[v]) * invTau;
    if ((cb == pcb) && (l16 == rloc)) posLds[rloc] = s;   // positive logit
    if ((cb == rb) && (l16 == rloc)) s = NEG_BIG;         // mask diagonal
    const float mo = m[v];
    const float mn = fmaxf(mo, s);
    l[v] = l[v] * __expf(mo - mn) + __expf(s - mn);       // online LSE
    m[v] = mn;
  }
}

// ------------------------------------------------- fused Gram + online LSE
__global__ __launch_bounds__(256) void simclr_kernel(const float* __restrict__ zn,
                                                     float* __restrict__ row_loss) {
  __shared__ __align__(16) float smemA[16 * DIM];  // 64 KB row strip (of 320 KB WGP LDS)
  __shared__ float posLds[16];
  __shared__ float mergeM[8 * 16];
  __shared__ float mergeL[8 * 16];

  const int rb   = blockIdx.x;                     // row block 0..255
  const int tid  = threadIdx.x;
  const int wave = tid >> 5;                       // 8 waves (wave32)
  const int lane = tid & 31;
  const int half = lane >> 4;                      // 0: K pair {0,1}, 1: {2,3}
  const int l16  = lane & 15;

  // Async-DMA the 16-row A strip into LDS: memory -> LDS directly (no VGPR
  // round trip), tracked by ASYNCcnt. 256 lanes * 16 B * 16 sweeps = 64 KB.
  {
    const char*    gsrc     = (const char*)(zn + (size_t)rb * 16 * DIM);
    const unsigned lds_base = (unsigned)(size_t)(void*)smemA;  // LDS offset = addr[31:0]
    #pragma unroll
    for (int it = 0; it < 16; ++it) {
      const unsigned loff = lds_base + (unsigned)(tid * 16 + it * 4096);
      const void*    g    = gsrc + tid * 16 + it * 4096;
      asm volatile("global_load_async_to_lds_b128 %0, %1, off"
                   :: "v"(loff), "v"(g) : "memory");
    }
    asm volatile("s_wait_asynccnt 0x0" ::: "memory");
  }
  __syncthreads();

  // A(16x4) and B(4x16) f32 fragments share the same per-lane addressing:
  // lane<16 holds K = 4t+{0,1}, lane>=16 holds K = 4t+{2,3}  -> one float2 each.
  const float* Aptr = smemA + l16 * DIM + half * 2;

  float m[8], l[8];
  #pragma unroll
  for (int v = 0; v < 8; ++v) { m[v] = NEG_BIG; l[v] = 0.f; }

  const int   pcb    = (rb + 128) & 255;           // partner (positive) column block
  const float invTau = 1.0f / TAU;

  // Each wave covers 32 column blocks {wave + 8k}; process them in pairs so the
  // A fragment feeds 2 WMMAs (1.5 loads per WMMA) with 4 independent acc chains.
  for (int j = 0; j < 16; ++j) {
    const int cb0 = wave + 16 * j;
    const int cb1 = cb0 + 8;
    const float* B0 = zn + (size_t)(cb0 * 16 + l16) * DIM + half * 2;
    const float* B1 = zn + (size_t)(cb1 * 16 + l16) * DIM + half * 2;
    v8f acc00 = {0.f, 0.f, 0.f, 0.f, 0.f, 0.f, 0.f, 0.f};
    v8f acc01 = {0.f, 0.f, 0.f, 0.f, 0.f, 0.f, 0.f, 0.f};
    v8f acc10 = {0.f, 0.f, 0.f, 0.f, 0.f, 0.f, 0.f, 0.f};
    v8f acc11 = {0.f, 0.f, 0.f, 0.f, 0.f, 0.f, 0.f, 0.f};
    #pragma unroll 4
    for (int t = 0; t < DIM / 4; t += 2) {
      v2f a0  = *(const v2f*)(Aptr + 4 * t);
      v2f a1  = *(const v2f*)(Aptr + 4 * t + 4);
      v2f b00 = *(const v2f*)(B0 + 4 * t);
      v2f b01 = *(const v2f*)(B0 + 4 * t + 4);
      v2f b10 = *(const v2f*)(B1 + 4 * t);
      v2f b11 = *(const v2f*)(B1 + 4 * t + 4);
      acc00 = __builtin_amdgcn_wmma_f32_16x16x4_f32(false, a0, false, b00,
                                                    (short)0, acc00, false, false);
      acc10 = __builtin_amdgcn_wmma_f32_16x16x4_f32(false, a0, false, b10,
                                                    (short)0, acc10, false, false);
      acc01 = __builtin_amdgcn_wmma_f32_16x16x4_f32(false, a1, false, b01,
                                                    (short)0, acc01, false, false);
      acc11 = __builtin_amdgcn_wmma_f32_16x16x4_f32(false, a1, false, b11,
                                                    (short)0, acc11, false, false);
    }
    tile_update(cb0, rb, pcb, invTau, acc00, acc01, l16, half, posLds, m, l);
    tile_update(cb1, rb, pcb, invTau, acc10, acc11, l16, half, posLds, m, l);
  }

  // Reduce (max, sumexp) across the 16 lanes of each half (wave32 butterflies).
  #pragma unroll
  for (int v = 0; v < 8; ++v) {
    #pragma unroll
    for (int off = 8; off >= 1; off >>= 1) {
      const float mo = __shfl_xor(m[v], off, 32);
      const float lo = __shfl_xor(l[v], off, 32);
      const float mn = fmaxf(m[v], mo);
      l[v] = l[v] * __expf(m[v] - mn) + lo * __expf(mo - mn);
      m[v] = mn;
    }
    if (l16 == 0) {
      const int rloc = v + 8 * half;
      mergeM[wave * 16 + rloc] = m[v];
      mergeL[wave * 16 + rloc] = l[v];
    }
  }
  __syncthreads();

  // Merge the 8 waves' partial LSEs and emit per-row loss.
  if (tid < 16) {
    float M = NEG_BIG, L = 0.f;
    #pragma unroll
    for (int w = 0; w < 8; ++w) {
      const float mo = mergeM[w * 16 + tid];
      const float lo = mergeL[w * 16 + tid];
      const float mn = fmaxf(M, mo);
      L = L * __expf(M - mn) + lo * __expf(mo - mn);
      M = mn;
    }
    row_loss[rb * 16 + tid] = M + __logf(L) - posLds[tid];
  }
}

// ---------------------------------------------------------------- final sum
__global__ __launch_bounds__(256) void reduce_kernel(const float* __restrict__ row_loss,
                                                     float* __restrict__ out) {
  __shared__ float red[8];
  const int tid = threadIdx.x;
  float s = 0.f;
  for (int i = tid; i < NROW; i += 256) s += row_loss[i];
  #pragma unroll
  for (int off = 16; off >= 1; off >>= 1) s += __shfl_xor(s, off, 32);
  if ((tid & 31) == 0) red[tid >> 5] = s;
  __syncthreads();
  if (tid == 0) {
    float t = 0.f;
    #pragma unroll
    for (int w = 0; w < 8; ++w) t += red[w];
    out[0] = t / (float)NROW;                      // poisoned d_out rewritten every call
  }
}

// ---------------------------------------------------------------------------
extern "C" void kernel_launch(void* const* d_in, const int* in_sizes, int n_in,
                              void* d_out, int out_size, void* d_ws, size_t ws_size,
                              hipStream_t stream) {
  const float* z_orig = (const float*)d_in[0];
  const float* z_aug  = (const float*)d_in[1];
  float* zn       = (float*)d_ws;                       // 4096*1024 f32 = 16 MB
  float* row_loss = zn + (size_t)NROW * DIM;            // + 16 KB

  nrm_kernel<<<NROW, 256, 0, stream>>>(z_orig, z_aug, zn);
  simclr_kernel<<<NBLK, 256, 0, stream>>>(zn, row_loss);
  reduce_kernel<<<1, 256, 0, stream>>>(row_loss, (float*)d_out);
}